// VariationalGCNEncoder_3470333575320
// MI455X (gfx1250) — compile-verified
//
#include <hip/hip_runtime.h>

// ---------------------------------------------------------------------------
// Variational GCN encoder for MI455X (gfx1250, wave32).
//  - Dense GEMMs via V_WMMA_F32_16X16X4_F32 (full fp32 precision; GEMM is
//    ~1.2 GFLOP total, negligible vs. the scatter phases).
//  - Edge aggregation via GLOBAL_ATOMIC_ADD_F32 (L2-resident: whole working
//    set ~40MB << 192MB L2; ~54M atomics per layer at 23.3 TB/s HBM /
//    L2 bandwidth is the true roofline).
// ---------------------------------------------------------------------------

typedef __attribute__((ext_vector_type(2))) float v2f;
typedef __attribute__((ext_vector_type(8))) float v8f;

#define N_NODES 50000
#define E_EDGES 800000
#define E_TOT   850000            // edges + self loops
#define F_IN    128
#define F_HID   64
#define F_CAT   64                // 32 (mu) + 32 (logstd), fused second GEMM

// ------------------------------ utility ------------------------------------

__global__ void zero_f32(float* __restrict__ p, int n) {
  int i = blockIdx.x * blockDim.x + threadIdx.x;
  if (i < n) p[i] = 0.0f;
}

// degree with self loops: deg[d] += 1 for every edge, + 1 per node (self loop)
__global__ void deg_kernel(const long long* __restrict__ ei, float* __restrict__ deg) {
  int e = blockIdx.x * blockDim.x + threadIdx.x;
  if (e >= E_TOT) return;
  int d = (e < E_EDGES) ? (int)ei[E_EDGES + e] : (e - E_EDGES);
  atomicAdd(&deg[d], 1.0f);
}

// deg -> D^{-1/2} in place (deg >= 1 always due to self loops, but guard)
__global__ void dinv_kernel(float* __restrict__ deg) {
  int i = blockIdx.x * blockDim.x + threadIdx.x;
  if (i >= N_NODES) return;
  float d = deg[i];
  deg[i] = (d > 0.0f) ? rsqrtf(d) : 0.0f;
}

// ------------------------------ WMMA GEMMs ---------------------------------
// One wave32 computes a 16x16 fp32 output tile with V_WMMA_F32_16X16X4_F32.
// A (16x4): lane l holds row l%16, K = v + 2*(l/16) in VGPR v.
// B (4x16): lane l holds col l%16, K = v + 2*(l/16) in VGPR v.
// D (16x16): VGPR r, lane l -> row r + 8*(l/16), col l%16.

// Y[N_NODES, 64] = X[N_NODES, 128] @ W[128, 64]
__global__ __launch_bounds__(256) void gemm1_wmma(const float* __restrict__ X,
                                                  const float* __restrict__ W,
                                                  float* __restrict__ Y) {
  int wave = (blockIdx.x * blockDim.x + threadIdx.x) >> 5;
  int lane = threadIdx.x & 31;
  const int tiles_n = F_HID / 16;               // 4
  int tm = wave / tiles_n;
  int tn = wave % tiles_n;
  if (tm >= N_NODES / 16) return;               // wave-uniform exit (EXEC all-1)

  int lmod = lane & 15;
  int khalf = (lane >> 4) * 2;                  // 0 or 2
  int row = tm * 16 + lmod;                     // A row for this lane
  int col = tn * 16 + lmod;                     // B col for this lane

  const float* xp = X + (long long)row * F_IN + khalf;
  const float* wp = W + (long long)khalf * F_HID + col;

  v8f acc = {};
  #pragma unroll 4
  for (int k0 = 0; k0 < F_IN; k0 += 4) {
    v2f a, b;
    a.x = xp[k0 + 0];                           // contiguous K pair
    a.y = xp[k0 + 1];
    b.x = wp[(long long)k0 * F_HID];
    b.y = wp[(long long)(k0 + 1) * F_HID];
    acc = __builtin_amdgcn_wmma_f32_16x16x4_f32(false, a, false, b,
                                                (short)0, acc, false, false);
  }

  int rbase = tm * 16 + (lane >> 4) * 8;
  #pragma unroll
  for (int r = 0; r < 8; ++r)
    Y[(long long)(rbase + r) * F_HID + col] = acc[r];
}

// Y[N_NODES, 64] = H[N_NODES, 64] @ [W_mu | W_logstd]   (each [64, 32])
__global__ __launch_bounds__(256) void gemm2_wmma(const float* __restrict__ H,
                                                  const float* __restrict__ Wmu,
                                                  const float* __restrict__ Wls,
                                                  float* __restrict__ Y) {
  int wave = (blockIdx.x * blockDim.x + threadIdx.x) >> 5;
  int lane = threadIdx.x & 31;
  const int tiles_n = F_CAT / 16;               // 4 (tiles 0-1: mu, 2-3: logstd)
  int tm = wave / tiles_n;
  int tn = wave % tiles_n;
  if (tm >= N_NODES / 16) return;

  int lmod = lane & 15;
  int khalf = (lane >> 4) * 2;
  int row = tm * 16 + lmod;

  const float* W = (tn < 2) ? Wmu : Wls;        // 16-wide tile never straddles
  int col_local = ((tn & 1) * 16) + lmod;       // column within the 32-wide W

  const float* hp = H + (long long)row * F_HID + khalf;
  const float* wp = W + (long long)khalf * 32 + col_local;

  v8f acc = {};
  #pragma unroll 4
  for (int k0 = 0; k0 < F_HID; k0 += 4) {
    v2f a, b;
    a.x = hp[k0 + 0];
    a.y = hp[k0 + 1];
    b.x = wp[(long long)k0 * 32];
    b.y = wp[(long long)(k0 + 1) * 32];
    acc = __builtin_amdgcn_wmma_f32_16x16x4_f32(false, a, false, b,
                                                (short)0, acc, false, false);
  }

  int rbase = tm * 16 + (lane >> 4) * 8;
  int col = tn * 16 + lmod;
  #pragma unroll
  for (int r = 0; r < 8; ++r)
    Y[(long long)(rbase + r) * F_CAT + col] = acc[r];
}

// ------------------------------ scatter ------------------------------------
// 16 threads per edge, float4 gather per thread -> fully coalesced 256B row
// reads; 4 scalar f32 atomics per thread to the destination row.

__global__ void scatter1(const float* __restrict__ Hsrc,     // [N, 64]
                         const long long* __restrict__ ei,
                         const float* __restrict__ dinv,
                         float* __restrict__ out) {          // [N, 64] accum
  long long t = (long long)blockIdx.x * blockDim.x + threadIdx.x;
  if (t >= (long long)E_TOT * 16) return;
  int e = (int)(t >> 4);
  int c = ((int)t & 15) * 4;
  int s, d;
  if (e < E_EDGES) { s = (int)ei[e]; d = (int)ei[E_EDGES + e]; }
  else             { s = d = e - E_EDGES; }
  float nrm = dinv[s] * dinv[d];
  const float4 hv = *(const float4*)(Hsrc + (long long)s * F_HID + c);
  float* o = out + (long long)d * F_HID + c;
  atomicAdd(o + 0, hv.x * nrm);
  atomicAdd(o + 1, hv.y * nrm);
  atomicAdd(o + 2, hv.z * nrm);
  atomicAdd(o + 3, hv.w * nrm);
}

// second layer: accumulate directly into d_out, split (mu, logstd) halves
__global__ void scatter2(const float* __restrict__ Hsrc,     // [N, 64] (mu|ls)
                         const long long* __restrict__ ei,
                         const float* __restrict__ dinv,
                         float* __restrict__ dout) {         // mu[N,32] ++ ls[N,32]
  long long t = (long long)blockIdx.x * blockDim.x + threadIdx.x;
  if (t >= (long long)E_TOT * 16) return;
  int e = (int)(t >> 4);
  int c = ((int)t & 15) * 4;                    // 0..60, never straddles 32
  int s, d;
  if (e < E_EDGES) { s = (int)ei[e]; d = (int)ei[E_EDGES + e]; }
  else             { s = d = e - E_EDGES; }
  float nrm = dinv[s] * dinv[d];
  const float4 hv = *(const float4*)(Hsrc + (long long)s * F_CAT + c);
  float* o = (c < 32)
      ? dout + (long long)d * 32 + c
      : dout + (long long)N_NODES * 32 + (long long)d * 32 + (c - 32);
  atomicAdd(o + 0, hv.x * nrm);
  atomicAdd(o + 1, hv.y * nrm);
  atomicAdd(o + 2, hv.z * nrm);
  atomicAdd(o + 3, hv.w * nrm);
}

// ------------------------------ epilogues ----------------------------------

__global__ void bias_relu(float* __restrict__ h, const float* __restrict__ b) {
  int i = blockIdx.x * blockDim.x + threadIdx.x;
  if (i >= N_NODES * F_HID) return;
  h[i] = fmaxf(h[i] + b[i & (F_HID - 1)], 0.0f);
}

// init d_out with broadcast biases (atomics accumulate on top)
__global__ void out_init(float* __restrict__ dout,
                         const float* __restrict__ bmu,
                         const float* __restrict__ bls) {
  int i = blockIdx.x * blockDim.x + threadIdx.x;
  if (i >= N_NODES * 64) return;
  int node = i >> 6, f = i & 63;
  if (f < 32) dout[(long long)node * 32 + f] = bmu[f];
  else        dout[(long long)N_NODES * 32 + (long long)node * 32 + (f - 32)] = bls[f - 32];
}

// ------------------------------ launch -------------------------------------

extern "C" void kernel_launch(void* const* d_in, const int* in_sizes, int n_in,
                              void* d_out, int out_size, void* d_ws, size_t ws_size,
                              hipStream_t stream) {
  const float*     x   = (const float*)d_in[0];
  const long long* ei  = (const long long*)d_in[1];   // [2, 800000] int64
  const float*     W1  = (const float*)d_in[2];
  const float*     b1  = (const float*)d_in[3];
  const float*     Wmu = (const float*)d_in[4];
  const float*     bmu = (const float*)d_in[5];
  const float*     Wls = (const float*)d_in[6];
  const float*     bls = (const float*)d_in[7];
  float* out = (float*)d_out;

  float* ws   = (float*)d_ws;
  float* deg  = ws;                                   // [50000] -> dinv in place
  float* bufA = ws + 50048;                           // [50000 x 64] GEMM out
  float* bufB = bufA + (size_t)N_NODES * 64;          // [50000 x 64] aggregation

  const int BT = 256;
  const long long sc_threads = (long long)E_TOT * 16;
  const unsigned sc_blocks = (unsigned)((sc_threads + BT - 1) / BT);
  const unsigned nf_blocks = (N_NODES * 64 + BT - 1) / BT;
  const unsigned gemm_blocks = ((N_NODES / 16) * 4 + 7) / 8;  // 8 waves/block

  // degrees -> D^{-1/2}
  zero_f32<<<(N_NODES + BT - 1) / BT, BT, 0, stream>>>(deg, N_NODES);
  deg_kernel<<<(E_TOT + BT - 1) / BT, BT, 0, stream>>>(ei, deg);
  dinv_kernel<<<(N_NODES + BT - 1) / BT, BT, 0, stream>>>(deg);

  // layer 1: h0 = x @ W1 ; agg ; relu(+b1)
  gemm1_wmma<<<gemm_blocks, BT, 0, stream>>>(x, W1, bufA);
  zero_f32<<<nf_blocks, BT, 0, stream>>>(bufB, N_NODES * 64);
  scatter1<<<sc_blocks, BT, 0, stream>>>(bufA, ei, deg, bufB);
  bias_relu<<<nf_blocks, BT, 0, stream>>>(bufB, b1);

  // layer 2 (fused mu|logstd): t = h1 @ [Wmu|Wls] ; agg into d_out
  gemm2_wmma<<<gemm_blocks, BT, 0, stream>>>(bufB, Wmu, Wls, bufA);
  out_init<<<nf_blocks, BT, 0, stream>>>(out, bmu, bls);
  scatter2<<<sc_blocks, BT, 0, stream>>>(bufA, ei, deg, out);
}